// PropGCNLayer_46308337386025
// MI455X (gfx1250) — compile-verified
//
#include <hip/hip_runtime.h>

#define N_NODES 50000
#define N_EDGES 800000
#define CH 64

typedef float v2f __attribute__((ext_vector_type(2)));
typedef float v8f __attribute__((ext_vector_type(8)));

// ---------------------------------------------------------------------------
// Kernel 1: weight reparameterization + KL reduction (single block).
//   w = mu + eps * exp(log_sigma)
//   kl = sum 0.5*(exp(2ls) + mu^2 - 2ls - 1)   (PRIOR_VAR = 1 -> log(pv)=0)
// Weights are stored K-pair interleaved for WMMA B tiles:
//   wp[(k>>1)*2*CH + n*2 + (k&1)] = w[k][n]
// so a B fragment {W[k][n], W[k+1][n]} is one aligned 8-byte load.
// ---------------------------------------------------------------------------
__global__ __launch_bounds__(256)
void weights_kl_kernel(const float* __restrict__ mu_m, const float* __restrict__ ls_m,
                       const float* __restrict__ ep_m,
                       const float* __restrict__ mu_s, const float* __restrict__ ls_s,
                       const float* __restrict__ ep_s,
                       float* __restrict__ wp_m, float* __restrict__ wp_s,
                       float* __restrict__ kl_out) {
  __shared__ float red[256];
  float acc = 0.0f;
  for (int i = threadIdx.x; i < CH * CH; i += 256) {
    const int k = i >> 6;           // IN_CH index (row of W)
    const int n = i & 63;           // OUT_CH index (col of W)
    const int pidx = (k >> 1) * (2 * CH) + n * 2 + (k & 1);

    float mu = mu_m[i], ls = ls_m[i];
    wp_m[pidx] = mu + ep_m[i] * expf(ls);
    acc += 0.5f * (expf(2.0f * ls) + mu * mu - 2.0f * ls - 1.0f);

    mu = mu_s[i]; ls = ls_s[i];
    wp_s[pidx] = mu + ep_s[i] * expf(ls);
    acc += 0.5f * (expf(2.0f * ls) + mu * mu - 2.0f * ls - 1.0f);
  }
  red[threadIdx.x] = acc;
  __syncthreads();
  for (int s = 128; s > 0; s >>= 1) {
    if (threadIdx.x < (unsigned)s) red[threadIdx.x] += red[threadIdx.x + s];
    __syncthreads();
  }
  if (threadIdx.x == 0) kl_out[0] = red[0];
}

// ---------------------------------------------------------------------------
// Kernel 2: zero the two accumulator regions of d_out (new_mean, log_var)
// ---------------------------------------------------------------------------
__global__ __launch_bounds__(256)
void zero_kernel(float4* __restrict__ p, int n4) {
  int i = blockIdx.x * 256 + threadIdx.x;
  if (i < n4) p[i] = make_float4(0.f, 0.f, 0.f, 0.f);
}

// ---------------------------------------------------------------------------
// Kernel 3: support = x @ W via V_WMMA_F32_16X16X4_F32.
// One wave32 per 16-node tile; both branches (mean, std^2) sequentially.
// A tile 16x4 fp32: lane<16 -> M=lane, K={kb,kb+1};  lane>=16 -> K={kb+2,kb+3}
//   -> contiguous float2 load per lane from x.
// B tile 4x16:      lane<16 -> N=lane, K={kb,kb+1};  lane>=16 -> K={kb+2,kb+3}
//   -> single float2 load per lane from K-pair-interleaved wp.
// C/D 16x16 fp32:   VGPR i -> row i (lanes 0-15) / row i+8 (lanes 16-31)
// ---------------------------------------------------------------------------
__global__ __launch_bounds__(256)
void gemm_wmma_kernel(const float* __restrict__ xmean, const float* __restrict__ xstd,
                      const float* __restrict__ wp_m, const float* __restrict__ wp_s,
                      float* __restrict__ sup_m, float* __restrict__ sup_v,
                      int nTiles) {
  const int wave = threadIdx.x >> 5;
  const int lane = threadIdx.x & 31;
  const int tile = blockIdx.x * (blockDim.x >> 5) + wave;
  if (tile >= nTiles) return;                 // wave-uniform: EXEC stays all-1s

  const int row0  = tile * 16;
  const int mrow  = lane & 15;                // A row / B col within tile
  const int khalf = (lane >> 4) << 1;         // 0 or 2

  for (int br = 0; br < 2; ++br) {
    const float* __restrict__ xb = br ? xstd : xmean;
    const float* __restrict__ wp = br ? wp_s : wp_m;
    float* __restrict__ sup      = br ? sup_v : sup_m;

    v8f acc[4];
    acc[0] = (v8f)0; acc[1] = (v8f)0; acc[2] = (v8f)0; acc[3] = (v8f)0;

    #pragma unroll 4
    for (int kb = 0; kb < CH; kb += 4) {
      const int k0 = kb + khalf;              // even
      v2f a = *(const v2f*)(xb + (size_t)(row0 + mrow) * CH + k0);
      if (br) { a.x = a.x * a.x; a.y = a.y * a.y; }   // var = std^2

      const float* bbase = wp + (size_t)(k0 >> 1) * (2 * CH);
      #pragma unroll
      for (int nt = 0; nt < 4; ++nt) {
        const int n = nt * 16 + mrow;
        v2f b = *(const v2f*)(bbase + n * 2);
        acc[nt] = __builtin_amdgcn_wmma_f32_16x16x4_f32(
            /*neg_a=*/false, a, /*neg_b=*/false, b,
            /*c_mod=*/(short)0, acc[nt], /*reuse_a=*/false, /*reuse_b=*/false);
      }
    }

    #pragma unroll
    for (int nt = 0; nt < 4; ++nt) {
      #pragma unroll
      for (int i = 0; i < 8; ++i) {
        const int r = (lane < 16) ? i : (i + 8);
        const int c = nt * 16 + mrow;
        sup[(size_t)(row0 + r) * CH + c] = acc[nt][i];
      }
    }
  }
}

// ---------------------------------------------------------------------------
// Kernel 4: edge scatter. 16 threads per edge, 4 channels (float4) each.
//   out_mean[dst]   += ew   * sup_m[src]
//   out_logvar[dst] += ew^2 * sup_v[src]
// ---------------------------------------------------------------------------
__global__ __launch_bounds__(256)
void scatter_kernel(const int* __restrict__ ei, const float* __restrict__ ew,
                    const float* __restrict__ sup_m, const float* __restrict__ sup_v,
                    float* __restrict__ out_m, float* __restrict__ out_lv,
                    int nEdges) {
  const int t   = blockIdx.x * 256 + threadIdx.x;
  const int e   = t >> 4;
  const int ch4 = (t & 15) << 2;
  if (e >= nEdges) return;

  const int src = ei[e];
  const int dst = ei[nEdges + e];
  const float w  = ew[e];
  const float w2 = w * w;

  const float4 m = *(const float4*)(sup_m + (size_t)src * CH + ch4);
  const float4 v = *(const float4*)(sup_v + (size_t)src * CH + ch4);

  float* om = out_m  + (size_t)dst * CH + ch4;
  float* ov = out_lv + (size_t)dst * CH + ch4;

  unsafeAtomicAdd(om + 0, w * m.x);
  unsafeAtomicAdd(om + 1, w * m.y);
  unsafeAtomicAdd(om + 2, w * m.z);
  unsafeAtomicAdd(om + 3, w * m.w);
  unsafeAtomicAdd(ov + 0, w2 * v.x);
  unsafeAtomicAdd(ov + 1, w2 * v.y);
  unsafeAtomicAdd(ov + 2, w2 * v.z);
  unsafeAtomicAdd(ov + 3, w2 * v.w);
}

// ---------------------------------------------------------------------------
// Kernel 5: new_std = sqrt(exp(log_var) + 1e-6), in place on std region
// ---------------------------------------------------------------------------
__global__ __launch_bounds__(256)
void finalize_kernel(float* __restrict__ lv, int n) {
  int i = blockIdx.x * 256 + threadIdx.x;
  if (i < n) lv[i] = sqrtf(expf(lv[i]) + 1e-6f);
}

// ---------------------------------------------------------------------------
extern "C" void kernel_launch(void* const* d_in, const int* in_sizes, int n_in,
                              void* d_out, int out_size, void* d_ws, size_t ws_size,
                              hipStream_t stream) {
  const float* mean = (const float*)d_in[0];
  const float* stdv = (const float*)d_in[1];
  const int*   ei   = (const int*)d_in[2];
  const float* ew   = (const float*)d_in[3];
  const float* mu_m = (const float*)d_in[4];
  const float* ls_m = (const float*)d_in[5];
  const float* ep_m = (const float*)d_in[6];
  const float* mu_s = (const float*)d_in[7];
  const float* ls_s = (const float*)d_in[8];
  const float* ep_s = (const float*)d_in[9];

  const int nNodes = in_sizes[0] / CH;          // 50000
  const int nEdges = in_sizes[2] / 2;           // 800000
  const int nFeat  = nNodes * CH;               // 3.2M

  float* out_mean = (float*)d_out;              // [nNodes, CH]
  float* out_std  = out_mean + nFeat;           // [nNodes, CH] (log_var accum, then std)
  float* out_kl   = out_std + nFeat;            // scalar

  // workspace layout (floats)
  float* wp_m  = (float*)d_ws;                  // 4096 (K-pair interleaved)
  float* wp_s  = wp_m + CH * CH;                // 4096
  float* sup_m = wp_s + CH * CH;                // nFeat
  float* sup_v = sup_m + nFeat;                 // nFeat

  // 1. weights (packed layout) + KL
  weights_kl_kernel<<<1, 256, 0, stream>>>(mu_m, ls_m, ep_m, mu_s, ls_s, ep_s,
                                           wp_m, wp_s, out_kl);

  // 2. zero accumulators (both output regions, 2*nFeat floats)
  const int n4 = (2 * nFeat) / 4;
  zero_kernel<<<(n4 + 255) / 256, 256, 0, stream>>>((float4*)d_out, n4);

  // 3. WMMA GEMMs: support = mean@Wm, support_v = std^2@Ws
  const int nTiles = nNodes / 16;               // 3125 (50000 % 16 == 0)
  const int wavesPerBlock = 8;                  // 256 threads
  gemm_wmma_kernel<<<(nTiles + wavesPerBlock - 1) / wavesPerBlock, 256, 0, stream>>>(
      mean, stdv, wp_m, wp_s, sup_m, sup_v, nTiles);

  // 4. edge scatter (16 threads/edge)
  const long long nThreads = (long long)nEdges * 16;
  scatter_kernel<<<(int)((nThreads + 255) / 256), 256, 0, stream>>>(
      ei, ew, sup_m, sup_v, out_mean, out_std, nEdges);

  // 5. finalize std
  finalize_kernel<<<(nFeat + 255) / 256, 256, 0, stream>>>(out_std, nFeat);
}